// FeedForwardModel_80376017977894
// MI455X (gfx1250) — compile-verified
//
#include <hip/hip_runtime.h>
#include <hip/hip_bf16.h>

typedef float v2f __attribute__((ext_vector_type(2)));
typedef float v8f __attribute__((ext_vector_type(8)));

#define NBATCH 4096
#define DIM    100
#define TSTEPS 50
#define DT     0.02f
#define EPS_BN 1e-6f
#define STP    112            // padded feature stride
#define WSTRIDE 114           // LDS weight stride (bank-conflict-free, even)
#define NCTILE 7              // 7 x 16 = 112 >= max outN (110)
#define STATS_STRIDE (8*STP)  // per-step: 4 layers x (sum[112], sumsq[112])
#define RCPN   (1.0f/4096.0f)

// ---------------------------------------------------------------------------
__global__ void zero_kernel(float* __restrict__ p, int n) {
    int i = blockIdx.x * 256 + threadIdx.x;
    if (i < n) p[i] = 0.0f;
}

// ---------------------------------------------------------------------------
// dw [N, DIM, T] -> dwT [T, N, DIM]  (into the xs output region; each step
// consumes its slice coalesced and then overwrites it with xs[t]).
__global__ void transpose_dw_kernel(const float* __restrict__ dw,
                                    float* __restrict__ dwT) {
    int idx = blockIdx.x * 256 + threadIdx.x;
    const int total = TSTEPS * NBATCH * DIM;
    if (idx >= total) return;
    int t   = idx / (NBATCH * DIM);
    int rem = idx - t * (NBATCH * DIM);
    int n   = rem / DIM;
    int d   = rem - n * DIM;
    dwT[idx] = dw[((size_t)n * DIM + d) * TSTEPS + t];
}

// ---------------------------------------------------------------------------
// Per-step elementwise kernel: one wave32 per batch row.
__global__ void step_kernel(int t, int isFinal,
                            float* __restrict__ X, float* __restrict__ Y,
                            const float* __restrict__ A3,
                            const float* __restrict__ stats3,
                            const float* __restrict__ y_init,
                            const float* __restrict__ z_init,
                            const float* __restrict__ bn3w,
                            const float* __restrict__ bn3b,
                            float* xs_dw,                 // aliased dwT / xs
                            float* __restrict__ us,
                            float* __restrict__ y_out,
                            float* __restrict__ ye_out,
                            float* __restrict__ stats0) {
    __shared__ float s_sum[DIM];
    __shared__ float s_sq[DIM];

    const int tid  = threadIdx.x;
    const int lane = tid & 31;
    const int wav  = tid >> 5;
    const int n    = blockIdx.x * 8 + wav;

    if (tid < DIM) { s_sum[tid] = 0.0f; s_sq[tid] = 0.0f; }
    __syncthreads();

    float xold[4], dwv[4];
    float px2 = 0.0f, pz = 0.0f, pzdw = 0.0f;

    const size_t slice = ((size_t)t * NBATCH + n) * DIM;

#pragma unroll
    for (int j = 0; j < 4; ++j) {
        int d = lane + 32 * j;
        xold[j] = 0.0f; dwv[j] = 0.0f;
        if (d < DIM) {
            float dwx = xs_dw[slice + d];
            float xo  = (t == 0) ? 0.0f : X[n * DIM + d];
            float z;
            if (t == 0) {
                z = z_init[d];
            } else {
                float m  = stats3[d] * RCPN;
                float v  = stats3[STP + d] * RCPN - m * m;
                float a3 = A3[n * STP + d];
                z = ((a3 - m) * rsqrtf(v + EPS_BN) * bn3w[d] + bn3b[d]) * 1e-4f;
            }
            px2  += xo * xo;
            pz   += z;
            pzdw += z * dwx;
            xold[j] = xo; dwv[j] = dwx;
        }
    }

#pragma unroll
    for (int off = 16; off > 0; off >>= 1) {
        px2  += __shfl_xor(px2,  off);
        pz   += __shfl_xor(pz,   off);
        pzdw += __shfl_xor(pzdw, off);
    }

    float u    = fminf(fmaxf(-pz, -1.0f), 1.0f);
    float yold = (t == 0) ? y_init[0] : Y[n];
    float ynew = yold - DT * 0.5f * (px2 + u * u) + DT * pz * u + pzdw;

#pragma unroll
    for (int j = 0; j < 4; ++j) {
        int d = lane + 32 * j;
        if (d < DIM) xs_dw[slice + d] = xold[j];
    }
    if (lane == 0) us[(size_t)t * NBATCH + n] = u;

    if (!isFinal) {
        if (lane == 0) Y[n] = ynew;
#pragma unroll
        for (int j = 0; j < 4; ++j) {
            int d = lane + 32 * j;
            if (d < DIM) {
                float xn = xold[j] * (1.0f - DT) + u * DT + dwv[j];
                X[n * DIM + d] = xn;
                atomicAdd(&s_sum[d], xn);
                atomicAdd(&s_sq[d],  xn * xn);
            }
        }
    } else {
        if (lane == 0) { y_out[n] = ynew; ye_out[n] = 0.5f * px2; }
    }
    __syncthreads();
    if (!isFinal && tid < DIM) {
        atomicAdd(&stats0[tid],       s_sum[tid]);
        atomicAdd(&stats0[STP + tid], s_sq[tid]);
    }
}

// ---------------------------------------------------------------------------
// Fused  BN(+ReLU) -> GEMM -> BN-stat accumulate, register-blocked:
// each wave owns a 16-row stripe and all 7 column tiles (7 accumulators),
// so one A fragment feeds 7 back-to-back v_wmma_f32_16x16x4_f32.
// W and the BN scale/shift are staged zero-padded in LDS -> branch-free
// inner loop. block = 128 (4 waves -> 64 rows), grid = M/64.
__global__ void gemm_bn_kernel(const float* __restrict__ in, int inStride, int K,
                               const float* __restrict__ stats_in,
                               const float* __restrict__ bnw,
                               const float* __restrict__ bnb,
                               int do_relu,
                               const float* __restrict__ W,    // [outN, K]
                               const float* __restrict__ bias, // nullable
                               float* __restrict__ out,        // stride STP
                               int outN,
                               float* __restrict__ stats_out) {
    __shared__ float  Wl[NCTILE * 16 * WSTRIDE];  // ~51 KB, zero-padded
    __shared__ float2 ss[STP];                    // (scale, shift) per k

    const int tid = threadIdx.x;
    const int Kp  = (K + 3) & ~3;

    // per-feature BN scale/shift (zero beyond K -> padded A elements = 0)
    for (int k = tid; k < STP; k += 128) {
        float sc = 0.0f, sh = 0.0f;
        if (k < K) {
            float m = stats_in[k] * RCPN;
            float v = stats_in[STP + k] * RCPN - m * m;
            sc = rsqrtf(v + EPS_BN) * bnw[k];
            sh = bnb[k] - m * sc;
        }
        ss[k] = make_float2(sc, sh);
    }
    // zero-padded weight tile: Wl[c * WSTRIDE + k] = W[c, k]
    for (int i = tid; i < NCTILE * 16 * WSTRIDE; i += 128) {
        int c = i / WSTRIDE;
        int k = i - c * WSTRIDE;
        float v = 0.0f;
        if (c < outN && k < K) v = W[(size_t)c * K + k];
        Wl[i] = v;
    }
    __syncthreads();

    const int lane  = threadIdx.x & 31;
    const int wave  = threadIdx.x >> 5;
    const int rowT  = blockIdx.x * 64 + wave * 16;
    const int cl    = lane & 15;
    const int khalf = (lane >> 4) << 1;           // 0 or 2
    const float* inRow = in + (size_t)(rowT + cl) * inStride;

    v8f acc[NCTILE];
#pragma unroll
    for (int j = 0; j < NCTILE; ++j) acc[j] = (v8f){};

    for (int kb = 0; kb < Kp; kb += 4) {
        int k0 = kb + khalf;
        // A pair (8B-aligned: rows even-strided, k0 even)
        float2 av = *(const float2*)(inRow + k0);
        float4 sv = *(const float4*)(&ss[k0]);    // sc0, sh0, sc1, sh1
        float a0 = fmaf(av.x, sv.x, sv.y);
        float a1 = fmaf(av.y, sv.z, sv.w);
        if (do_relu) { a0 = fmaxf(a0, 0.0f); a1 = fmaxf(a1, 0.0f); }
        v2f a; a.x = a0; a.y = a1;

#pragma unroll
        for (int j = 0; j < NCTILE; ++j) {
            const float2 bw = *(const float2*)(&Wl[(j * 16 + cl) * WSTRIDE + k0]);
            v2f b; b.x = bw.x; b.y = bw.y;
            acc[j] = __builtin_amdgcn_wmma_f32_16x16x4_f32(
                false, a, false, b, (short)0, acc[j], false, false);
        }
    }

    // Epilogue: bias, store, per-feature stats (sum & sumsq)
    const int rbas = rowT + ((lane >> 4) << 3);
#pragma unroll
    for (int j = 0; j < NCTILE; ++j) {
        int col = j * 16 + cl;
        if (col < outN) {
            float bv   = bias ? bias[col] : 0.0f;
            float lsum = 0.0f, lsq = 0.0f;
#pragma unroll
            for (int i = 0; i < 8; ++i) {
                float val = acc[j][i] + bv;
                out[(size_t)(rbas + i) * STP + col] = val;
                lsum += val;
                lsq  += val * val;
            }
            atomicAdd(&stats_out[col],       lsum);
            atomicAdd(&stats_out[STP + col], lsq);
        }
    }
}

// ---------------------------------------------------------------------------
__global__ void loss_kernel(const float* __restrict__ y,
                            const float* __restrict__ ye,
                            float* __restrict__ loss_out) {
    __shared__ float r1[256];
    __shared__ float r2[256];
    int tid = threadIdx.x;
    float s1 = 0.0f, s2 = 0.0f;
    for (int i = tid; i < NBATCH; i += 256) {
        float d  = y[i] - ye[i];
        float ad = fabsf(d);
        s1 += (ad < 1.0f) ? 0.5f * d * d : (ad - 0.5f);
        s2 += ye[i];
    }
    r1[tid] = s1; r2[tid] = s2;
    __syncthreads();
    for (int off = 128; off > 0; off >>= 1) {
        if (tid < off) { r1[tid] += r1[tid + off]; r2[tid] += r2[tid + off]; }
        __syncthreads();
    }
    if (tid == 0) {
        float sl1 = r1[0] * RCPN;
        float my  = r2[0] * RCPN;
        loss_out[0] = 100.0f * (sl1 + my * my);   // 2 * DELTA_CLIP = 100
    }
}

// ---------------------------------------------------------------------------
extern "C" void kernel_launch(void* const* d_in, const int* in_sizes, int n_in,
                              void* d_out, int out_size, void* d_ws, size_t ws_size,
                              hipStream_t stream) {
    const float* dw     = (const float*)d_in[0];
    const float* y_init = (const float*)d_in[1];
    const float* z_init = (const float*)d_in[2];
    const float* bn0w   = (const float*)d_in[3];
    const float* bn0b   = (const float*)d_in[4];
    const float* w1     = (const float*)d_in[5];
    const float* bn1w   = (const float*)d_in[6];
    const float* bn1b   = (const float*)d_in[7];
    const float* w2     = (const float*)d_in[8];
    const float* bn2w   = (const float*)d_in[9];
    const float* bn2b   = (const float*)d_in[10];
    const float* w3     = (const float*)d_in[11];
    const float* b3     = (const float*)d_in[12];
    const float* bn3w   = (const float*)d_in[13];
    const float* bn3b   = (const float*)d_in[14];

    float* out     = (float*)d_out;
    float* loss_o  = out;
    float* y_o     = out + 1;
    float* ye_o    = out + 1 + NBATCH;
    float* xs      = out + 1 + 2 * NBATCH;                       // [T,N,DIM]
    float* us      = xs + (size_t)TSTEPS * NBATCH * DIM;         // [T,N]

    float* w   = (float*)d_ws;
    float* X   = w;                            // 4096*100
    float* Y   = X + NBATCH * DIM;             // 4096
    float* A1  = Y + NBATCH;                   // 4096*112
    float* A2  = A1 + NBATCH * STP;            // 4096*112
    float* A3  = A2 + NBATCH * STP;            // 4096*112
    float* STS = A3 + NBATCH * STP;            // 50 * 896

    const int statsN = TSTEPS * STATS_STRIDE;
    zero_kernel<<<(statsN + 255) / 256, 256, 0, stream>>>(STS, statsN);

    const int totalDW = TSTEPS * NBATCH * DIM;
    transpose_dw_kernel<<<(totalDW + 255) / 256, 256, 0, stream>>>(dw, xs);

    dim3 gGemm(NBATCH / 64);   // 64 blocks; each wave covers all 7 col tiles

    for (int t = 0; t < TSTEPS - 1; ++t) {
        float* st   = STS + t * STATS_STRIDE;
        const float* stz = (t > 0) ? (STS + (t - 1) * STATS_STRIDE + 6 * STP)
                                   : nullptr;
        step_kernel<<<NBATCH / 8, 256, 0, stream>>>(
            t, 0, X, Y, A3, stz, y_init, z_init, bn3w, bn3b,
            xs, us, y_o, ye_o, st);

        // layer1: bn0(x) @ w1^T         -> A1 (+bn1 stats)
        gemm_bn_kernel<<<gGemm, 128, 0, stream>>>(
            X, DIM, 100, st, bn0w, bn0b, 0, w1, nullptr, A1, 110, st + 2 * STP);
        // layer2: relu(bn1(A1)) @ w2^T  -> A2 (+bn2 stats)
        gemm_bn_kernel<<<gGemm, 128, 0, stream>>>(
            A1, STP, 110, st + 2 * STP, bn1w, bn1b, 1, w2, nullptr, A2, 110,
            st + 4 * STP);
        // layer3: relu(bn2(A2)) @ w3^T + b3 -> A3 (+bn3 stats)
        gemm_bn_kernel<<<gGemm, 128, 0, stream>>>(
            A2, STP, 110, st + 4 * STP, bn2w, bn2b, 1, w3, b3, A3, 100,
            st + 6 * STP);
    }

    step_kernel<<<NBATCH / 8, 256, 0, stream>>>(
        TSTEPS - 1, 1, X, Y, A3, STS + (TSTEPS - 2) * STATS_STRIDE + 6 * STP,
        y_init, z_init, bn3w, bn3b, xs, us, y_o, ye_o,
        STS + (TSTEPS - 1) * STATS_STRIDE);

    loss_kernel<<<1, 256, 0, stream>>>(y_o, ye_o, loss_o);
}